// IntraAgg_54511724921157
// MI455X (gfx1250) — compile-verified
//
#include <hip/hip_runtime.h>
#include <stdint.h>

// ---- vector types for WMMA / TDM builtins -------------------------------
typedef __attribute__((ext_vector_type(2))) float v2f;
typedef __attribute__((ext_vector_type(8))) float v8f;
typedef __attribute__((ext_vector_type(4))) unsigned int u32x4;
typedef __attribute__((ext_vector_type(8))) int i32x8;
typedef __attribute__((ext_vector_type(4))) int i32x4;

#define FEAT 64            // embedding dim
#define KNEI 32            // max neighbors
#define NODES_PER_WG 16    // one WMMA M-tile of batch nodes per workgroup
#define ROW_DW 72          // 64 data dwords + 8 pad dwords (TDM pad => conflict-free LDS)
#define ROWS_PER_WG (NODES_PER_WG * KNEI)  // 512 staged rows, 144 KB of the 320 KB WGP LDS

__global__ __launch_bounds__(128) void intra_agg_kernel(
    const float* __restrict__ features,
    const int*   __restrict__ neigh_idx,
    const int*   __restrict__ neigh_counts,
    float*       __restrict__ out,
    unsigned int n_nodes)
{
    __shared__ float smem[ROWS_PER_WG * ROW_DW];   // 147456 bytes

    const int tid  = threadIdx.x;
    const int lane = tid & 31;
    const int wv   = tid >> 5;          // wave id 0..3 : owns N-tile cols [16*wv, 16*wv+16)
    const int m    = lane & 15;         // matrix row (M) / column-within-tile index
    const int half = lane >> 4;         // lane half selects K pairs {0,1} vs {2,3}
    const int nodeBase = blockIdx.x * NODES_PER_WG;

    // per-lane: neighbor count of batch node (nodeBase + m)
    const int   cnt = neigh_counts[nodeBase + m];
    const float inv = 1.0f / (float)cnt;

    // =====================================================================
    // Stage 1: TDM gather-mode DMA — 8 indexed embedding rows per descriptor
    // D# layout per CDNA5 ISA
    // =====================================================================
    const uint64_t gaddr   = (uint64_t)(uintptr_t)features;
    const uint32_t ldsBase = (uint32_t)(uintptr_t)&smem[0];  // flat->LDS: low 32 bits

    u32x4 g0;
    g0.x = 0xC0000001u;                                   // count=1 | gather_index_size=32b | gather_mode
    g0.y = 0u;                                            // lds_addr (per-descriptor below)
    g0.z = (uint32_t)gaddr;                               // global_addr[31:0]
    g0.w = ((uint32_t)(gaddr >> 32) & 0x01FFFFFFu)        // global_addr[56:32]
         | (2u << 30);                                    // type=2 ("image")

    i32x8 g1;
    g1.s0 = (int)0x0F520000u;    // data_size=4B | pad_enable | pad_interval=64dw | pad_amount=8dw
    g1.s1 = (int)(64u << 16);                             // tensor_dim0 = 64 (bits 79:48, lo16)
    g1.s2 = (int)((n_nodes & 0xFFFFu) << 16);             // dim0 hi16=0 | tensor_dim1 lo16
    g1.s3 = (int)(((n_nodes >> 16) & 0xFFFFu)             // tensor_dim1 hi16
         | (64u << 16));                                  // tile_dim0 = 64 elements/row
    g1.s4 = 8;                                            // tile_dim1 = 8 valid gather indices
    g1.s5 = 64;                                           // tensor_dim0_stride = 64
    g1.s6 = 0;
    g1.s7 = 0;

    const i32x8 gx = {0, 0, 0, 0, 0, 0, 0, 0};            // extra operand (6-arg toolchain), zero-filled

    #pragma unroll 1
    for (int t = 0; t < 16; ++t) {                        // wave wv gathers nodes 4*wv..4*wv+3
        const int lb = wv * 4 + (t >> 2);                 // local node
        const int g  = t & 3;                             // neighbor octet
        const int* ip = neigh_idx + (size_t)(nodeBase + lb) * KNEI + g * 8;
        i32x4 g2, g3;                                     // 32-bit row indices 0..7
        g2.x = __builtin_amdgcn_readfirstlane(ip[0]);
        g2.y = __builtin_amdgcn_readfirstlane(ip[1]);
        g2.z = __builtin_amdgcn_readfirstlane(ip[2]);
        g2.w = __builtin_amdgcn_readfirstlane(ip[3]);
        g3.x = __builtin_amdgcn_readfirstlane(ip[4]);
        g3.y = __builtin_amdgcn_readfirstlane(ip[5]);
        g3.z = __builtin_amdgcn_readfirstlane(ip[6]);
        g3.w = __builtin_amdgcn_readfirstlane(ip[7]);
        g0.y = ldsBase + (uint32_t)(lb * KNEI + g * 8) * (ROW_DW * 4u);
        __builtin_amdgcn_tensor_load_to_lds(g0, g1, g2, g3, gx, 0);
    }

    __builtin_amdgcn_s_wait_tensorcnt(0);   // own TDM ops landed in LDS
    __syncthreads();                        // all waves' rows visible

    // =====================================================================
    // Stage 2: D = A(block-diag weights) x B(gathered rows), fp32 WMMA
    // K-chunk c covers global k = 4c..4c+3 inside node b = c>>3's 32-row segment
    // A layout (ISA §7.12.2): vgpr0 = K{0 | 2}, vgpr1 = K{1 | 3} by lane half
    // =====================================================================
    v8f acc = {};
    #pragma unroll 4
    for (int c = 0; c < 128; ++c) {
        const int b  = c >> 3;
        const int j0 = ((c & 7) << 2) + half * 2;         // neighbor index of this K pair
        v2f A;
        A.x = (m == b && j0     < cnt) ? inv : 0.0f;
        A.y = (m == b && j0 + 1 < cnt) ? inv : 0.0f;
        const float* rp = &smem[(c * 4 + half * 2) * ROW_DW + wv * 16 + m];
        v2f B;
        B.x = rp[0];          // row k = 4c + 2*half
        B.y = rp[ROW_DW];     // row k = 4c + 2*half + 1
        acc = __builtin_amdgcn_wmma_f32_16x16x4_f32(
            false, A, false, B, (short)0, acc, false, false);
    }

    // C/D layout: vgpr r holds M = r (lanes 0-15) / r+8 (lanes 16-31), N = lane&15
    float* op = out + (size_t)nodeBase * FEAT + wv * 16 + m;
    #pragma unroll
    for (int r = 0; r < 8; ++r) {
        const int M = r + half * 8;
        op[(size_t)M * FEAT] = fmaxf(acc[r], 0.0f);       // ReLU
    }
}

extern "C" void kernel_launch(void* const* d_in, const int* in_sizes, int n_in,
                              void* d_out, int out_size, void* d_ws, size_t ws_size,
                              hipStream_t stream) {
    const float* features     = (const float*)d_in[0];
    const int*   neigh_idx    = (const int*)d_in[1];
    const int*   neigh_counts = (const int*)d_in[2];
    float*       out          = (float*)d_out;

    const unsigned int n_nodes = (unsigned int)(in_sizes[0] / FEAT);
    const int batch = in_sizes[2];

    dim3 grid(batch / NODES_PER_WG);
    dim3 block(128);
    hipLaunchKernelGGL(intra_agg_kernel, grid, block, 0, stream,
                       features, neigh_idx, neigh_counts, out, n_nodes);
}